// MLPRegressor_19456201851335
// MI455X (gfx1250) — compile-verified
//
#include <hip/hip_runtime.h>

// ---------------------------------------------------------------------------
// Fused 15-layer MLP (128->128->128->64->64->32->32->16->16->8->8->4->4->2->2->1)
// MI455X / gfx1250, wave32, V_WMMA_F32_16X16X4_F32 (exact fp32 math).
//
// Key insight: with M=16 per wave and the f32 WMMA A/D layouts, the whole
// activation pipeline is WAVE-PRIVATE (A reads row lane%16, D writes rows
// g+8*half of the same strip). So each wave runs as an independent persistent
// worker over 16-row batch strips with ZERO block barriers in the main loop.
// Weights (zero-padded to 16-multiples) live in LDS, loaded once per block.
// ---------------------------------------------------------------------------

typedef __attribute__((ext_vector_type(2))) float v2f;
typedef __attribute__((ext_vector_type(8))) float v8f;

#define MLP_BATCH   524288
#define STRIP       16
#define NUM_STRIPS  (MLP_BATCH / STRIP)    // 32768
#define NWAVES      6
#define THREADS     (NWAVES * 32)          // 192
#define NBLOCKS     1024
#define ACT_STRIDE  132                    // 128+4: conflict-free, 16B-aligned rows
#define ACT_ELEMS   (STRIP * ACT_STRIDE)   // 2112 floats per ping-pong buffer

// Padded weights in LDS, row stride = Kpad+2 (b64 loads hit distinct banks).
#define W_TOTAL 51872
#define B_TOTAL 592

struct Params {
    const float* W[15];
    const float* B[15];
};

__device__ __forceinline__ void wave_fence() {
    // Compiler-ordering fence; per-wave DS pipeline is in-order in HW.
    __builtin_amdgcn_fence(__ATOMIC_ACQ_REL, "wavefront");
}

// Cooperative zero-padded weight preload into LDS.
template<int FANIN, int FANOUT, int KPAD, int NPAD>
__device__ __forceinline__ void load_w(float* dst, const float* __restrict__ src, int tid) {
    constexpr int WS = KPAD + 2;
    for (int i = tid; i < NPAD * WS; i += THREADS) {
        const int n = i / WS;
        const int k = i - n * WS;
        float v = 0.0f;
        if (n < FANOUT && k < FANIN) v = src[n * FANIN + k];
        dst[i] = v;
    }
}

template<int FANOUT, int NPAD>
__device__ __forceinline__ void load_b(float* dst, const float* __restrict__ src, int tid) {
    for (int i = tid; i < NPAD; i += THREADS) dst[i] = (i < FANOUT) ? src[i] : 0.0f;
}

// One layer on a wave-private 16-row strip.
// A layout (16x4 f32): lane = M%16, K = v + 2*(lane>>4)     [ISA 7.12.2]
// B layout (4x16 f32): N = lane%16, K = v + 2*(lane>>4)
// D layout (16x16 f32): M = g + 8*(lane>>4), N = lane%16
// NB n-tiles are blocked together: one A fragment feeds NB independent
// accumulator chains (ILP) and A-side LDS traffic is divided by NB.
template<int K, int N, bool RELU>
__device__ __forceinline__ void mlp_layer(const float* __restrict__ aIn,
                                          float* __restrict__ aOut,
                                          const float* __restrict__ wl,
                                          const float* __restrict__ bl,
                                          int lr, int half) {
    constexpr int WS  = K + 2;
    constexpr int NTT = N / 16;
    constexpr int NB  = (NTT % 4 == 0) ? 4 : ((NTT % 2 == 0) ? 2 : 1);
    const float* arow = aIn + lr * ACT_STRIDE + 2 * half;

#pragma unroll
    for (int ntb = 0; ntb < NTT; ntb += NB) {
        v8f acc[NB];
#pragma unroll
        for (int j = 0; j < NB; ++j)
            acc[j] = v8f{0.f, 0.f, 0.f, 0.f, 0.f, 0.f, 0.f, 0.f};

#pragma unroll
        for (int kc = 0; kc < K / 4; ++kc) {
            const v2f a = *reinterpret_cast<const v2f*>(arow + 4 * kc);
#pragma unroll
            for (int j = 0; j < NB; ++j) {
                const float* brow = wl + ((ntb + j) * 16 + lr) * WS + 2 * half;
                const v2f b = *reinterpret_cast<const v2f*>(brow + 4 * kc);
                acc[j] = __builtin_amdgcn_wmma_f32_16x16x4_f32(
                    false, a, false, b, (short)0, acc[j], false, false);
            }
        }
#pragma unroll
        for (int j = 0; j < NB; ++j) {
            const int n0 = (ntb + j) * 16 + lr;
            const float bias = bl[n0];
#pragma unroll
            for (int g = 0; g < 8; ++g) {
                float v = acc[j][g] + bias;
                if (RELU) v = fmaxf(v, 0.0f);
                aOut[(g + 8 * half) * ACT_STRIDE + n0] = v;
            }
        }
    }
}

// Final layer (Kpad=16 -> 1 feature): lanes with n==0 store 16 rows to global.
__device__ __forceinline__ void mlp_last(const float* __restrict__ aIn,
                                         const float* __restrict__ wl,
                                         const float* __restrict__ bl,
                                         float* __restrict__ out, int rowBase,
                                         int lr, int half) {
    constexpr int K = 16, WS = K + 2;
    const float* arow = aIn + lr * ACT_STRIDE + 2 * half;
    const float* brow = wl + lr * WS + 2 * half;
    v8f acc = {0.f, 0.f, 0.f, 0.f, 0.f, 0.f, 0.f, 0.f};
#pragma unroll
    for (int kc = 0; kc < K / 4; ++kc) {
        const v2f a = *reinterpret_cast<const v2f*>(arow + 4 * kc);
        const v2f b = *reinterpret_cast<const v2f*>(brow + 4 * kc);
        acc = __builtin_amdgcn_wmma_f32_16x16x4_f32(
            false, a, false, b, (short)0, acc, false, false);
    }
    if (lr == 0) {
        const float bias = bl[0];
#pragma unroll
        for (int g = 0; g < 8; ++g)
            out[rowBase + g + 8 * half] = acc[g] + bias;
    }
}

__global__ __launch_bounds__(THREADS, 1)
void mlp_fused_wmma(const float* __restrict__ x, float* __restrict__ out, Params p) {
    __shared__ float s_w[W_TOTAL];
    __shared__ float s_b[B_TOTAL];
    __shared__ float s_act[NWAVES][2][ACT_ELEMS];   // wave-private ping-pong

    const int tid  = threadIdx.x;
    const int lane = tid & 31;
    const int lr   = lane & 15;
    const int half = lane >> 4;
    const int w    = tid >> 5;

    // ---- one-time weight/bias preload (only block-wide sync in the kernel) ----
    load_w<128, 128, 128, 128>(s_w + 0,     p.W[0],  tid);
    load_w<128, 128, 128, 128>(s_w + 16640, p.W[1],  tid);
    load_w<128,  64, 128,  64>(s_w + 33280, p.W[2],  tid);
    load_w< 64,  64,  64,  64>(s_w + 41600, p.W[3],  tid);
    load_w< 64,  32,  64,  32>(s_w + 45824, p.W[4],  tid);
    load_w< 32,  32,  32,  32>(s_w + 47936, p.W[5],  tid);
    load_w< 32,  16,  32,  16>(s_w + 49024, p.W[6],  tid);
    load_w< 16,  16,  16,  16>(s_w + 49568, p.W[7],  tid);
    load_w< 16,   8,  16,  16>(s_w + 49856, p.W[8],  tid);
    load_w<  8,   8,  16,  16>(s_w + 50144, p.W[9],  tid);
    load_w<  8,   4,  16,  16>(s_w + 50432, p.W[10], tid);
    load_w<  4,   4,  16,  16>(s_w + 50720, p.W[11], tid);
    load_w<  4,   2,  16,  16>(s_w + 51008, p.W[12], tid);
    load_w<  2,   2,  16,  16>(s_w + 51296, p.W[13], tid);
    load_w<  2,   1,  16,  16>(s_w + 51584, p.W[14], tid);

    load_b<128, 128>(s_b + 0,   p.B[0],  tid);
    load_b<128, 128>(s_b + 128, p.B[1],  tid);
    load_b< 64,  64>(s_b + 256, p.B[2],  tid);
    load_b< 64,  64>(s_b + 320, p.B[3],  tid);
    load_b< 32,  32>(s_b + 384, p.B[4],  tid);
    load_b< 32,  32>(s_b + 416, p.B[5],  tid);
    load_b< 16,  16>(s_b + 448, p.B[6],  tid);
    load_b< 16,  16>(s_b + 464, p.B[7],  tid);
    load_b<  8,  16>(s_b + 480, p.B[8],  tid);
    load_b<  8,  16>(s_b + 496, p.B[9],  tid);
    load_b<  4,  16>(s_b + 512, p.B[10], tid);
    load_b<  4,  16>(s_b + 528, p.B[11], tid);
    load_b<  2,  16>(s_b + 544, p.B[12], tid);
    load_b<  2,  16>(s_b + 560, p.B[13], tid);
    load_b<  1,  16>(s_b + 576, p.B[14], tid);
    __syncthreads();

    float* actA = &s_act[w][0][0];
    float* actB = &s_act[w][1][0];

    // ---- barrier-free persistent loop: each wave walks 16-row strips ----
    const int waveStride = (int)gridDim.x * NWAVES;
    for (int s = (int)blockIdx.x * NWAVES + w; s < NUM_STRIPS; s += waveStride) {
        // Load this wave's 16x128 X strip (512B coalesced bursts per row).
        const float4* xs = reinterpret_cast<const float4*>(x) + (size_t)s * (STRIP * 32);
#pragma unroll
        for (int i = 0; i < STRIP; ++i)
            *reinterpret_cast<float4*>(&actA[i * ACT_STRIDE + lane * 4]) = xs[i * 32 + lane];

        // Prefetch this wave's next strip (8 KB) into the near cache.
        if (s + waveStride < NUM_STRIPS) {
            const char* nx = reinterpret_cast<const char*>(xs + (size_t)waveStride * (STRIP * 32));
            __builtin_prefetch(nx + lane * 256, 0, 3);
            __builtin_prefetch(nx + lane * 256 + 128, 0, 3);
        }
        wave_fence();

        mlp_layer<128, 128, true>(actA, actB, s_w + 0,     s_b + 0,   lr, half); wave_fence();
        mlp_layer<128, 128, true>(actB, actA, s_w + 16640, s_b + 128, lr, half); wave_fence();
        mlp_layer<128,  64, true>(actA, actB, s_w + 33280, s_b + 256, lr, half); wave_fence();
        mlp_layer< 64,  64, true>(actB, actA, s_w + 41600, s_b + 320, lr, half); wave_fence();
        mlp_layer< 64,  32, true>(actA, actB, s_w + 45824, s_b + 384, lr, half); wave_fence();
        mlp_layer< 32,  32, true>(actB, actA, s_w + 47936, s_b + 416, lr, half); wave_fence();
        mlp_layer< 32,  16, true>(actA, actB, s_w + 49024, s_b + 448, lr, half); wave_fence();
        mlp_layer< 16,  16, true>(actB, actA, s_w + 49568, s_b + 464, lr, half); wave_fence();
        mlp_layer< 16,  16, true>(actA, actB, s_w + 49856, s_b + 480, lr, half); wave_fence();
        mlp_layer< 16,  16, true>(actB, actA, s_w + 50144, s_b + 496, lr, half); wave_fence();
        mlp_layer< 16,  16, true>(actA, actB, s_w + 50432, s_b + 512, lr, half); wave_fence();
        mlp_layer< 16,  16, true>(actB, actA, s_w + 50720, s_b + 528, lr, half); wave_fence();
        mlp_layer< 16,  16, true>(actA, actB, s_w + 51008, s_b + 544, lr, half); wave_fence();
        mlp_layer< 16,  16, true>(actB, actA, s_w + 51296, s_b + 560, lr, half); wave_fence();
        mlp_last(actA, s_w + 51584, s_b + 576, out, s * STRIP, lr, half);
        wave_fence();
    }
}

extern "C" void kernel_launch(void* const* d_in, const int* in_sizes, int n_in,
                              void* d_out, int out_size, void* d_ws, size_t ws_size,
                              hipStream_t stream) {
    (void)in_sizes; (void)n_in; (void)out_size; (void)d_ws; (void)ws_size;

    Params p;
    for (int l = 0; l < 15; ++l) {
        p.W[l] = (const float*)d_in[1 + 2 * l];
        p.B[l] = (const float*)d_in[2 + 2 * l];
    }
    const float* x = (const float*)d_in[0];
    float* out = (float*)d_out;

    // 1024 persistent blocks x 6 independent waves = 6144 workers over 32768
    // strips; each block loads weights into LDS once (~210 MB L2-resident total).
    mlp_fused_wmma<<<NBLOCKS, THREADS, 0, stream>>>(x, out, p);
}